// Assigner_58308476010541
// MI455X (gfx1250) — compile-verified
//
#include <hip/hip_runtime.h>
#include <stdint.h>

#define BS   16
#define NB   64
#define NA   8400
#define NC   80
#define TOPK 13

// ---------------- CDNA5 async global->LDS helpers (gfx1250) ----------------
__device__ __forceinline__ void async_b128_to_lds(uint32_t lds_off, uint64_t gaddr) {
  asm volatile("global_load_async_to_lds_b128 %0, %1, off" :: "v"(lds_off), "v"(gaddr) : "memory");
}
__device__ __forceinline__ void async_b32_to_lds(uint32_t lds_off, uint64_t gaddr) {
  asm volatile("global_load_async_to_lds_b32 %0, %1, off" :: "v"(lds_off), "v"(gaddr) : "memory");
}
__device__ __forceinline__ void wait_async_zero() {
  asm volatile("s_wait_asynccnt 0" ::: "memory");
}

// ---------------- Pass A: masks, CIoU overlaps, align metric ----------------
__global__ __launch_bounds__(256) void pass_a(
    const float* __restrict__ pd_scores, const float* __restrict__ pd_bboxes,
    const float* __restrict__ anc,       const int*   __restrict__ gt_labels,
    const float* __restrict__ gt_bboxes, const float* __restrict__ mask_gt,
    float* __restrict__ w_align, float* __restrict__ w_over,
    unsigned* __restrict__ inlo, unsigned* __restrict__ inhi,
    unsigned* __restrict__ poslo, unsigned* __restrict__ poshi)
{
  __shared__ float sgt[NB * 4];
  __shared__ int   slab[NB];
  __shared__ float smgt[NB];
  __shared__ float satan1[NB];

  const int tid = threadIdx.x;
  const int b   = blockIdx.y;
  const int a   = blockIdx.x * 256 + tid;
  const bool valid = (a < NA);

  if (tid < NB) {
    async_b128_to_lds((uint32_t)(uintptr_t)&sgt[tid * 4],
                      (uint64_t)(uintptr_t)&gt_bboxes[((size_t)b * NB + tid) * 4]);
    async_b32_to_lds((uint32_t)(uintptr_t)&slab[tid],
                     (uint64_t)(uintptr_t)&gt_labels[(size_t)b * NB + tid]);
    async_b32_to_lds((uint32_t)(uintptr_t)&smgt[tid],
                     (uint64_t)(uintptr_t)&mask_gt[(size_t)b * NB + tid]);
  }
  wait_async_zero();
  __syncthreads();

  if (tid < NB) {
    float w1 = sgt[tid * 4 + 2] - sgt[tid * 4 + 0];
    float h1 = sgt[tid * 4 + 3] - sgt[tid * 4 + 1] + 1e-7f;
    satan1[tid] = atanf(w1 / h1);
  }
  __syncthreads();

  float px1 = 0.f, py1 = 0.f, px2 = 0.f, py2 = 0.f, ax = 0.f, ay = 0.f;
  if (valid) {
    __builtin_prefetch(pd_scores + ((size_t)b * NA + a) * NC, 0, 0);
    const float4 p = *(const float4*)&pd_bboxes[((size_t)b * NA + a) * 4];
    px1 = p.x; py1 = p.y; px2 = p.z; py2 = p.w;
    ax = anc[(size_t)a * 2 + 0];
    ay = anc[(size_t)a * 2 + 1];
  }
  const float w2 = px2 - px1;
  const float h2 = py2 - py1 + 1e-7f;
  const float atan2v = atanf(w2 / h2);
  const float parea  = w2 * h2;
  const float K = 4.0f / (3.14159265358979323846f * 3.14159265358979323846f);

  unsigned mlo = 0u, mhi = 0u;
  for (int j = 0; j < NB; ++j) {
    const float gx1 = sgt[j * 4 + 0], gy1 = sgt[j * 4 + 1];
    const float gx2 = sgt[j * 4 + 2], gy2 = sgt[j * 4 + 3];
    const bool in_gt = (ax - gx1 > 1e-9f) && (ay - gy1 > 1e-9f) &&
                       (gx2 - ax > 1e-9f) && (gy2 - ay > 1e-9f);
    const bool gt_on = (smgt[j] > 0.f);
    const bool tm = valid && in_gt && gt_on;

    float ov = 0.f, al = 0.f;
    if (tm) {
      const float w1 = gx2 - gx1;
      const float h1 = gy2 - gy1 + 1e-7f;
      float iw = fminf(gx2, px2) - fmaxf(gx1, px1); iw = fmaxf(iw, 0.f);
      float ih = fminf(gy2, py2) - fmaxf(gy1, py1); ih = fmaxf(ih, 0.f);
      const float inter = iw * ih;
      const float uni   = w1 * h1 + parea - inter + 1e-7f;
      const float iou   = inter / uni;
      const float cw = fmaxf(gx2, px2) - fminf(gx1, px1);
      const float ch = fmaxf(gy2, py2) - fminf(gy1, py1);
      const float c2 = cw * cw + ch * ch + 1e-7f;
      const float dx = px1 + px2 - gx1 - gx2;
      const float dy = py1 + py2 - gy1 - gy2;
      const float rho2 = (dx * dx + dy * dy) * 0.25f;
      const float dif = atan2v - satan1[j];
      const float v = K * dif * dif;
      const float alpha = v / (v - iou + 1.0f + 1e-7f);
      const float ciou = iou - (rho2 / c2 + v * alpha);
      ov = fmaxf(ciou, 0.f);
      const float s = pd_scores[((size_t)b * NA + a) * NC + slab[j]];
      const float o2 = ov * ov;
      al = s * (o2 * o2 * o2);  // score^1 * overlap^6
    }
    if (valid) {
      const size_t off = ((size_t)(b * NB + j)) * NA + a;
      w_align[off] = al;
      w_over[off]  = ov;
      if (in_gt && gt_on) {
        if (j < 32) mlo |= (1u << j); else mhi |= (1u << (j - 32));
      }
    }
  }
  if (valid) {
    const size_t idx = (size_t)b * NA + a;
    inlo[idx] = mlo;  inhi[idx] = mhi;
    poslo[idx] = 0u;  poshi[idx] = 0u;
  }
}

// ---------------- Pass B: exact top-13 per (b,gt) row -----------------------
__global__ __launch_bounds__(256) void pass_b(
    const float* __restrict__ w_align, const float* __restrict__ mask_gt,
    int* __restrict__ tk)
{
  const int row = blockIdx.x;     // b*NB + j
  const int tid = threadIdx.x;

  if (mask_gt[row] <= 0.f) {      // invalid row contributes nothing
    if (tid == 0) tk[row * TOPK] = -1;
    return;
  }

  __shared__ float sv[NA];
  __shared__ float rv[256];
  __shared__ int   ri[256];

  const float* rowp = w_align + (size_t)row * NA;
  for (int i = tid; i < NA; i += 256) sv[i] = rowp[i];
  __syncthreads();

  for (int k = 0; k < TOPK; ++k) {
    float bv = -2.f; int bi = 0x7fffffff;
    for (int i = tid; i < NA; i += 256) {
      const float v = sv[i];
      if (v > bv || (v == bv && i < bi)) { bv = v; bi = i; }
    }
    rv[tid] = bv; ri[tid] = bi;
    __syncthreads();
    for (int s = 128; s > 0; s >>= 1) {
      if (tid < s) {
        const float ov = rv[tid + s]; const int oi = ri[tid + s];
        if (ov > rv[tid] || (ov == rv[tid] && oi < ri[tid])) { rv[tid] = ov; ri[tid] = oi; }
      }
      __syncthreads();
    }
    if (tid == 0) { tk[row * TOPK + k] = ri[0]; sv[ri[0]] = -1.f; }
    __syncthreads();
  }
}

// ---------------- Pass C: scatter assignment bits, zero row maxima ----------
__global__ __launch_bounds__(256) void pass_c(
    const int* __restrict__ tk,
    const unsigned* __restrict__ inlo, const unsigned* __restrict__ inhi,
    unsigned* __restrict__ poslo, unsigned* __restrict__ poshi,
    unsigned* __restrict__ posal, unsigned* __restrict__ posov)
{
  const int t = blockIdx.x * blockDim.x + threadIdx.x;
  if (t >= BS * NB) return;
  posal[t] = 0u; posov[t] = 0u;
  if (tk[t * TOPK] < 0) return;
  const int b = t / NB, j = t % NB;
  const unsigned bit = (j < 32) ? (1u << j) : (1u << (j - 32));
  for (int k = 0; k < TOPK; ++k) {
    const int a = tk[t * TOPK + k];
    const size_t idx = (size_t)b * NA + a;
    const unsigned inm = (j < 32) ? inlo[idx] : inhi[idx];
    if (inm & bit) {
      if (j < 32) atomicOr(&poslo[idx], bit); else atomicOr(&poshi[idx], bit);
    }
  }
}

// ---------------- Pass D: multi-assign fix, fg/tgi, row maxima --------------
__global__ __launch_bounds__(256) void pass_d(
    const float* __restrict__ w_over, const float* __restrict__ w_align,
    const unsigned* __restrict__ poslo, const unsigned* __restrict__ poshi,
    unsigned* __restrict__ posal, unsigned* __restrict__ posov,
    float* __restrict__ anc_al, int* __restrict__ tgi_ws,
    float* __restrict__ out_fg, float* __restrict__ out_tgi)
{
  const int b = blockIdx.y;
  const int a = blockIdx.x * 256 + threadIdx.x;
  if (a >= NA) return;
  const size_t idx = (size_t)b * NA + a;
  const unsigned lo = poslo[idx], hi = poshi[idx];
  const int fg = __popc(lo) + __popc(hi);

  int js = -1;
  if (fg > 1) {
    float bv = -1e30f; int bj = 0;
    for (int j = 0; j < NB; ++j) {
      const float v = w_over[((size_t)(b * NB + j)) * NA + a];
      if (v > bv) { bv = v; bj = j; }          // first max wins (matches argmax)
    }
    js = bj;
  } else if (fg == 1) {
    js = (lo != 0u) ? __builtin_ctz(lo) : (32 + __builtin_ctz(hi));
  }

  float fgf = 0.f, aval = 0.f; int t = 0;
  if (js >= 0) {
    t = js; fgf = 1.f;
    const size_t off = ((size_t)(b * NB + js)) * NA + a;
    aval = w_align[off];
    const float ovv = w_over[off];
    atomicMax(&posal[b * NB + js], __float_as_uint(aval));  // values >= 0
    atomicMax(&posov[b * NB + js], __float_as_uint(ovv));
  }
  anc_al[idx] = aval;
  tgi_ws[idx] = t;
  out_fg[idx]  = fgf;
  out_tgi[idx] = (float)t;
}

// ---------------- Pass F: targets --------------------------------------------
__global__ __launch_bounds__(256) void pass_f(
    const int* __restrict__ tgi_ws, const float* __restrict__ anc_al,
    const unsigned* __restrict__ posal, const unsigned* __restrict__ posov,
    const int* __restrict__ gt_labels, const float* __restrict__ gt_bboxes,
    const float* __restrict__ out_fg,
    float* __restrict__ out_bbox, float* __restrict__ out_scores)
{
  const int b = blockIdx.y;
  const int a = blockIdx.x * 256 + threadIdx.x;
  if (a >= NA) return;
  const size_t idx = (size_t)b * NA + a;
  const int t = tgi_ws[idx];
  const bool fg = (out_fg[idx] > 0.5f);

  float norm = 0.f;
  if (fg) {
    const float pa = __uint_as_float(posal[b * NB + t]);
    const float po = __uint_as_float(posov[b * NB + t]);
    norm = anc_al[idx] * po / (pa + 1e-9f);
  }
  const int lab = gt_labels[(size_t)b * NB + t];
  const float4 gb = *(const float4*)&gt_bboxes[((size_t)b * NB + t) * 4];
  *(float4*)&out_bbox[idx * 4] = gb;

  float* sc = out_scores + idx * NC;
  for (int c = 0; c < NC; c += 4) {
    float4 z = make_float4(0.f, 0.f, 0.f, 0.f);
    if (fg) {
      if      (lab == c)     z.x = norm;
      else if (lab == c + 1) z.y = norm;
      else if (lab == c + 2) z.z = norm;
      else if (lab == c + 3) z.w = norm;
    }
    *(float4*)&sc[c] = z;
  }
}

extern "C" void kernel_launch(void* const* d_in, const int* in_sizes, int n_in,
                              void* d_out, int out_size, void* d_ws, size_t ws_size,
                              hipStream_t stream) {
  const float* pd_scores = (const float*)d_in[0];
  const float* pd_bboxes = (const float*)d_in[1];
  const float* anc       = (const float*)d_in[2];
  const int*   gt_labels = (const int*)  d_in[3];
  const float* gt_bboxes = (const float*)d_in[4];
  const float* mask_gt   = (const float*)d_in[5];

  // workspace carve (element counts)
  const size_t N_PAIR = (size_t)BS * NB * NA;   // 8,601,600
  const size_t N_ANC  = (size_t)BS * NA;        // 134,400
  float*    w_align = (float*)d_ws;
  float*    w_over  = w_align + N_PAIR;
  unsigned* inlo    = (unsigned*)(w_over + N_PAIR);
  unsigned* inhi    = inlo + N_ANC;
  unsigned* poslo   = inhi + N_ANC;
  unsigned* poshi   = poslo + N_ANC;
  int*      tk      = (int*)(poshi + N_ANC);    // BS*NB*TOPK
  unsigned* posal   = (unsigned*)(tk + (size_t)BS * NB * TOPK);
  unsigned* posov   = posal + (size_t)BS * NB;
  float*    anc_al  = (float*)(posov + (size_t)BS * NB);
  int*      tgi_ws  = (int*)(anc_al + N_ANC);

  // output carve (floats), tuple order: bboxes, scores, fg_mask, target_gt_idx
  float* out        = (float*)d_out;
  float* out_bbox   = out;                                  // 16*8400*4
  float* out_scores = out + (size_t)BS * NA * 4;            // 16*8400*80
  float* out_fg     = out_scores + (size_t)BS * NA * NC;    // 16*8400
  float* out_tgi    = out_fg + N_ANC;                       // 16*8400

  const dim3 gAnchor((NA + 255) / 256, BS);

  pass_a<<<gAnchor, 256, 0, stream>>>(pd_scores, pd_bboxes, anc, gt_labels,
                                      gt_bboxes, mask_gt,
                                      w_align, w_over, inlo, inhi, poslo, poshi);
  pass_b<<<BS * NB, 256, 0, stream>>>(w_align, mask_gt, tk);
  pass_c<<<(BS * NB + 255) / 256, 256, 0, stream>>>(tk, inlo, inhi,
                                                    poslo, poshi, posal, posov);
  pass_d<<<gAnchor, 256, 0, stream>>>(w_over, w_align, poslo, poshi,
                                      posal, posov, anc_al, tgi_ws,
                                      out_fg, out_tgi);
  pass_f<<<gAnchor, 256, 0, stream>>>(tgi_ws, anc_al, posal, posov,
                                      gt_labels, gt_bboxes, out_fg,
                                      out_bbox, out_scores);
}